// S4D_17197049053646
// MI455X (gfx1250) — compile-verified
//
#include <hip/hip_runtime.h>
#include <math.h>

// ---------------- problem constants ----------------
#define Hc   256
#define N2c  32
#define Bc   8
#define Lc   4096
#define NBL  (Bc * Lc)          // 32768 GEMM columns

typedef __attribute__((ext_vector_type(2))) float v2f;
typedef __attribute__((ext_vector_type(8))) float v8f;

// =====================================================================
// Kernel A: S4D state-space scan, fused D-skip + exact GELU.
// One wave32 per (b,h) sequence; lane n = complex state n (N2 == 32).
// Per 32-step chunk: 32 serial state updates (per-lane contribution kept
// in a register per step), then ONE transposing butterfly reduction
// (31 swizzles) that leaves y[l0+lane] in lane `lane`.
// Output stored transposed: y_t[h][b*L + l]  (B-matrix for the GEMM).
// =====================================================================
__global__ void __launch_bounds__(256)
s4d_scan_gelu(const float* __restrict__ u,
              const float* __restrict__ log_dt,
              const float* __restrict__ C,
              const float* __restrict__ log_A_real,
              const float* __restrict__ A_imag,
              const float* __restrict__ D,
              float* __restrict__ y_t)
{
    const int lane = threadIdx.x & 31;
    const int wv   = blockIdx.x * 8 + (threadIdx.x >> 5); // 0..B*H-1
    const int b    = wv >> 8;                             // / H
    const int h    = wv & (Hc - 1);

    // ---- per-(h,n) constants: w = exp(dt*A), Ct' = 2*C*(w-1)/A ----
    const int   hn = h * N2c + lane;
    const float dt = __expf(log_dt[h]);
    const float Ar = -__expf(log_A_real[hn]);
    const float Ai = A_imag[hn];
    const float er = __expf(Ar * dt);
    const float wr = er * __cosf(Ai * dt);
    const float wi = er * __sinf(Ai * dt);
    const float e1r = wr - 1.0f, e1i = wi;                // exp(dtA)-1
    const float inv = 1.0f / (Ar * Ar + Ai * Ai);
    const float dr  = (e1r * Ar + e1i * Ai) * inv;        // (w-1)/A
    const float di  = (e1i * Ar - e1r * Ai) * inv;
    const float Ccr = C[2 * hn + 0], Cci = C[2 * hn + 1];
    const float Ctr = 2.0f * (Ccr * dr - Cci * di);
    const float Cti = 2.0f * (Ccr * di + Cci * dr);
    const float Dh  = D[h];

    const float* up = u   + ((size_t)b * Hc + h) * Lc;
    float*       yp = y_t + (size_t)h * NBL + (size_t)b * Lc;

    float sr = 0.0f, si = 0.0f;
    for (int l0 = 0; l0 < Lc; l0 += 32) {
        const float uv = up[l0 + lane];      // 32 inputs, coalesced
        float c[32];                         // c[j] = this state's contrib @ step j
#pragma unroll
        for (int j = 0; j < 32; ++j) {
            // uniform broadcast of u[l0+j] via scalar readlane
            const float ul = __int_as_float(
                __builtin_amdgcn_readlane(__float_as_int(uv), j));
            const float nsr = fmaf(wr, sr, fmaf(-wi, si, ul));
            const float nsi = fmaf(wr, si, wi * sr);
            sr = nsr; si = nsi;
            c[j] = fmaf(Ctr, sr, -Cti * si);
        }
        // Transposing butterfly: after level `bit`, c[r] holds the partial
        // sum for step j = r + (lane & processed_bits) over partner states.
        // End state: c[0] = sum over all 32 states for step l0 + lane.
#pragma unroll
        for (int bit = 16; bit >= 1; bit >>= 1) {
            const bool hi = (lane & bit) != 0;
#pragma unroll
            for (int r = 0; r < bit; ++r) {
                const float send = hi ? c[r] : c[r + bit];
                const float recv = __shfl_xor(send, bit, 32);
                const float keep = hi ? c[r + bit] : c[r];
                c[r] = keep + recv;
            }
        }
        const float yk = fmaf(Dh, uv, c[0]);             // + D skip (step l0+lane)
        yp[l0 + lane] = 0.5f * yk * (1.0f + erff(yk * 0.70710678118654752f));
    }
}

// =====================================================================
// Kernel B: z = W(512x256) @ y_t(256x32768) + b, then GLU.
// Full f32 via V_WMMA_F32_16X16X4_F32.  Block = 8 waves:
//   gridDim.y -> a-rows o0..o0+15 (paired with g-rows o0+256..),
//   gridDim.x * 128 + warp*16 -> 16-column tile per wave.
// W tiles staged in LDS via CDNA5 async global->LDS copies (ASYNCcnt).
// =====================================================================
#define WP 260   // padded LDS row pitch (floats), keeps b64/b128 align

__global__ void __launch_bounds__(256)
s4d_gemm_glu(const float* __restrict__ W,
             const float* __restrict__ bias,
             const float* __restrict__ y_t,
             float* __restrict__ out)
{
    __shared__ __attribute__((aligned(16))) float sW[2][16 * WP]; // 33 KB

    const int tid = threadIdx.x;
    const int o0  = blockIdx.y * 16;

    // ---- async stage of 32 W rows (16 'a' + 16 'g'), 2048 float4 ----
    // LDS byte offset = low 32 bits of the LDS-aperture flat address.
    const unsigned ldsbase = (unsigned)(uintptr_t)&sW[0][0];
#pragma unroll
    for (int i = 0; i < 8; ++i) {
        const int q    = tid + i * 256;     // float4 index 0..2047
        const int row  = q >> 6;            // 0..31 (64 float4 per row)
        const int c4   = q & 63;
        const int orow = (row < 16) ? (o0 + row) : (256 + o0 + (row - 16));
        const float* src = W + (size_t)orow * Hc + c4 * 4;
        const unsigned dst =
            ldsbase + 4u * ((unsigned)(row >> 4) * (16 * WP) +
                            (unsigned)(row & 15) * WP + (unsigned)c4 * 4);
        asm volatile("global_load_async_to_lds_b128 %0, %1, off"
                     :: "v"(dst), "v"(src) : "memory");
    }
    asm volatile("s_wait_asynccnt 0x0" ::: "memory");
    __syncthreads();

    const int lane = tid & 31;
    const int warp = tid >> 5;
    const int half = lane >> 4;             // 0: K+0/1, 1: K+2/3 (ISA A/B layout)
    const int m    = lane & 15;
    const int col  = blockIdx.x * 128 + warp * 16 + m;   // GEMM column (b*L+l)

    const float* Wa = &sW[0][m * WP];
    const float* Wg = &sW[1][m * WP];
    const float* yb = y_t + col;

    v8f acc_a = {};
    v8f acc_g = {};

#pragma unroll 4
    for (int k0 = 0; k0 < Hc; k0 += 4) {
        const int kk = k0 + half * 2;
        const v2f Aa = *(const v2f*)(Wa + kk);           // A: 16x4 f32 tile
        const v2f Ag = *(const v2f*)(Wg + kk);
        v2f Bv;                                          // B: 4x16 f32 tile
        Bv.x = yb[(size_t)kk * NBL];
        Bv.y = yb[(size_t)(kk + 1) * NBL];
        acc_a = __builtin_amdgcn_wmma_f32_16x16x4_f32(false, Aa, false, Bv,
                                                      (short)0, acc_a, false, false);
        acc_g = __builtin_amdgcn_wmma_f32_16x16x4_f32(false, Ag, false, Bv,
                                                      (short)0, acc_g, false, false);
    }

    // epilogue: bias + GLU, write out[b][o][l]
    const int bidx = col >> 12;             // / L
    const int l    = col & (Lc - 1);
#pragma unroll
    for (int r = 0; r < 8; ++r) {
        const int   row = r + half * 8;     // C/D layout: vgpr r -> M=r(+8)
        const float a   = acc_a[r] + bias[o0 + row];
        const float g   = acc_g[r] + bias[256 + o0 + row];
        const float s   = 1.0f / (1.0f + __expf(-g));
        out[((size_t)bidx * Hc + (o0 + row)) * Lc + l] = a * s;
    }
}

// =====================================================================
extern "C" void kernel_launch(void* const* d_in, const int* in_sizes, int n_in,
                              void* d_out, int out_size, void* d_ws, size_t ws_size,
                              hipStream_t stream)
{
    const float* u          = (const float*)d_in[0];
    const float* log_dt     = (const float*)d_in[1];
    const float* C          = (const float*)d_in[2];
    const float* log_A_real = (const float*)d_in[3];
    const float* A_imag     = (const float*)d_in[4];
    const float* D          = (const float*)d_in[5];
    const float* W          = (const float*)d_in[6];
    const float* b          = (const float*)d_in[7];
    float*       out        = (float*)d_out;
    float*       y_t        = (float*)d_ws;   // H x (B*L) f32 = 33.5 MB

    // Kernel A: B*H = 2048 waves, 8 waves/block
    s4d_scan_gelu<<<dim3(256), dim3(256), 0, stream>>>(
        u, log_dt, C, log_A_real, A_imag, D, y_t);

    // Kernel B: 256 column-blocks x 16 row-tiles
    s4d_gemm_glu<<<dim3(NBL / 128, Hc / 16), dim3(256), 0, stream>>>(
        W, b, y_t, out);
}